// PGT_GConvLSTM_25890062860561
// MI455X (gfx1250) — compile-verified
//
#include <hip/hip_runtime.h>
#include <hip/hip_bf16.h>
#include <cmath>

typedef float v2f __attribute__((ext_vector_type(2)));
typedef float v8f __attribute__((ext_vector_type(8)));

__device__ __forceinline__ float sigmoidf_(float x) {
    return 1.0f / (1.0f + __expf(-x));
}

// One wave32 handles a 16-node tile. Gates computed as four 16x16 fp32 WMMA
// accumulators via V_WMMA_F32_16X16X4_F32 (K=8 for x@Wx, K=16 for h@Wh).
__global__ __launch_bounds__(256) void PGT_GConvLSTM_fused_kernel(
    const float* __restrict__ x,      // (N,8)
    const float* __restrict__ h,      // (N,16)
    const float* __restrict__ c,      // (N,16)
    const float* __restrict__ Wx,     // (8,64)
    const float* __restrict__ bx,     // (64)
    const float* __restrict__ Wh,     // (16,64)
    const float* __restrict__ bh,     // (64)
    const float* __restrict__ w_peep, // (3,16)
    const float* __restrict__ b_gates,// (4,16)
    const float* __restrict__ W_lin,  // (16,1)
    const float* __restrict__ b_lin,  // (1)
    float* __restrict__ out,          // (N)
    float* __restrict__ h_new,        // (N,16)
    float* __restrict__ c_new,        // (N,16)
    int n)
{
    const int lane = threadIdx.x & 31;
    const int wave = threadIdx.x >> 5;
    const long tile = (long)blockIdx.x * (blockDim.x >> 5) + wave;
    const long base = tile * 16;
    if (base >= n) return;              // wave-uniform: EXEC stays all-ones

    const int col = lane & 15;          // feature / B,C,D column
    const int hi  = lane >> 4;          // wave half

    // Per-feature constants (broadcast across the 8 accumulator rows)
    const float wp0  = w_peep[0 * 16 + col];
    const float wp1  = w_peep[1 * 16 + col];
    const float wp2  = w_peep[2 * 16 + col];
    const float wlin = W_lin[col];
    const float blin = b_lin[0];

    // Seed accumulators with the fused bias: bx + bh + b_gates
    v8f acc[4];
#pragma unroll
    for (int g = 0; g < 4; ++g) {
        const float bias = bx[g * 16 + col] + bh[g * 16 + col] + b_gates[g * 16 + col];
#pragma unroll
        for (int v = 0; v < 8; ++v) acc[g][v] = bias;
    }

    // ---- x @ Wx : K = 8 -> two 16x16x4 fp32 WMMA steps per gate ----
    // A layout: lane row = col; VGPR v holds K = kbase + v + 2*hi
    {
        const long rowA = (base + col) < (long)n ? (base + col) : (long)n - 1;
        const float* xrow = x + rowA * 8;
#pragma unroll
        for (int k = 0; k < 2; ++k) {
            v2f a;
            a.x = xrow[4 * k + 2 * hi + 0];
            a.y = xrow[4 * k + 2 * hi + 1];
#pragma unroll
            for (int g = 0; g < 4; ++g) {
                v2f b;
                b.x = Wx[(4 * k + 2 * hi + 0) * 64 + g * 16 + col];
                b.y = Wx[(4 * k + 2 * hi + 1) * 64 + g * 16 + col];
                acc[g] = __builtin_amdgcn_wmma_f32_16x16x4_f32(
                    /*neg_a=*/false, a, /*neg_b=*/false, b,
                    /*c_mod=*/(short)0, acc[g], /*reuse_a=*/false, /*reuse_b=*/false);
            }
        }
    }

    // ---- h @ Wh : K = 16 -> four 16x16x4 fp32 WMMA steps per gate ----
    {
        const long rowA = (base + col) < (long)n ? (base + col) : (long)n - 1;
        const float* hrow = h + rowA * 16;
#pragma unroll
        for (int k = 0; k < 4; ++k) {
            v2f a;
            a.x = hrow[4 * k + 2 * hi + 0];
            a.y = hrow[4 * k + 2 * hi + 1];
#pragma unroll
            for (int g = 0; g < 4; ++g) {
                v2f b;
                b.x = Wh[(4 * k + 2 * hi + 0) * 64 + g * 16 + col];
                b.y = Wh[(4 * k + 2 * hi + 1) * 64 + g * 16 + col];
                acc[g] = __builtin_amdgcn_wmma_f32_16x16x4_f32(
                    false, a, false, b, (short)0, acc[g], false, false);
            }
        }
    }

    // ---- Pointwise LSTM math in the C/D register layout ----
    // C/D layout: VGPR v, lane l -> node row = v + 8*hi, feature = col.
#pragma unroll
    for (int v = 0; v < 8; ++v) {
        const long row = base + v + 8 * hi;
        const bool ok = row < (long)n;
        const long rclamp = ok ? row : (long)n - 1;
        const float cv = c[rclamp * 16 + col];

        const float gi = acc[0][v];
        const float gf = acc[1][v];
        const float gc = acc[2][v];
        const float go = acc[3][v];

        const float ig = sigmoidf_(gi + wp0 * cv);
        const float fg = sigmoidf_(gf + wp1 * cv);
        const float tg = tanhf(gc);
        const float cn = fg * cv + ig * tg;
        const float og = sigmoidf_(go + wp2 * cn);
        const float hn = og * tanhf(cn);

        if (ok) {
            c_new[row * 16 + col] = cn;
            h_new[row * 16 + col] = hn;
        }

        // 16->1 output projection: relu(h_new) . W_lin, reduced across the
        // 16 lanes of each wave half via XOR shuffles (bit4 untouched).
        float p = fmaxf(hn, 0.0f) * wlin;
        p += __shfl_xor(p, 1, 32);
        p += __shfl_xor(p, 2, 32);
        p += __shfl_xor(p, 4, 32);
        p += __shfl_xor(p, 8, 32);
        if (col == 0 && ok) out[row] = p + blin;
    }
}

extern "C" void kernel_launch(void* const* d_in, const int* in_sizes, int n_in,
                              void* d_out, int out_size, void* d_ws, size_t ws_size,
                              hipStream_t stream) {
    (void)n_in; (void)out_size; (void)d_ws; (void)ws_size;
    const float* x       = (const float*)d_in[0];
    // d_in[1] = edge_index (unused), d_in[2] = edge_attr (unused)
    const float* h       = (const float*)d_in[3];
    const float* c       = (const float*)d_in[4];
    const float* Wx      = (const float*)d_in[5];
    const float* bx      = (const float*)d_in[6];
    const float* Wh      = (const float*)d_in[7];
    const float* bh      = (const float*)d_in[8];
    const float* w_peep  = (const float*)d_in[9];
    const float* b_gates = (const float*)d_in[10];
    const float* W_lin   = (const float*)d_in[11];
    const float* b_lin   = (const float*)d_in[12];

    const int n = in_sizes[0] / 8;     // N (x is N x 8)

    float* out   = (float*)d_out;                  // (N)
    float* h_new = out + (size_t)n;                // (N,16)
    float* c_new = h_new + (size_t)n * 16;         // (N,16)

    const int tiles = (n + 15) / 16;               // one wave32 per 16 nodes
    const int wavesPerBlock = 8;                   // 256 threads
    const int blocks = (tiles + wavesPerBlock - 1) / wavesPerBlock;

    PGT_GConvLSTM_fused_kernel<<<blocks, 256, 0, stream>>>(
        x, h, c, Wx, bx, Wh, bh, w_peep, b_gates, W_lin, b_lin,
        out, h_new, c_new, n);
}